// Attention_89180700934627
// MI455X (gfx1250) — compile-verified
//
#include <hip/hip_runtime.h>
#include <hip/hip_bf16.h>

// CDNA5 / gfx1250: wave32, WMMA f32 16x16x4 for full-precision small GEMMs.

typedef __attribute__((ext_vector_type(2))) float v2f;
typedef __attribute__((ext_vector_type(8))) float v8f;

// ---------------------------------------------------------------------------
// WMMA fp32 GEMM: C[M,N] = A[M,K] @ B + optional bias[N]
//   BT=true : B stored row-major [N,K]  (C = A @ B^T)
//   BT=false: B stored row-major [K,N]  (C = A @ B)
// One wave (32 threads) computes one 16x16 tile of C via V_WMMA_F32_16X16X4_F32.
// Fragment layouts per CDNA5 ISA 7.12.2 (32-bit A 16x4, 32-bit C/D 16x16).
// ---------------------------------------------------------------------------
template <bool BT, bool BIAS>
__launch_bounds__(32)
__global__ void wmma_gemm_f32(const float* __restrict__ A,
                              const float* __restrict__ B,
                              const float* __restrict__ bias,
                              float* __restrict__ C,
                              int M, int N, int K) {
    const int lane   = threadIdx.x & 31;
    const int mlocal = lane & 15;     // row within tile (A), col within tile (B/C)
    const int khalf  = lane >> 4;     // 0: K={0,1}, 1: K={2,3}
    const int row0   = blockIdx.y * 16;
    const int col0   = blockIdx.x * 16;

    const float* Arow = A + (size_t)(row0 + mlocal) * K + 2 * khalf;
    v8f acc = {};

    if (BT) {
        const float* Brow = B + (size_t)(col0 + mlocal) * K + 2 * khalf;
#pragma unroll 4
        for (int k = 0; k < K; k += 4) {
            v2f a = *(const v2f*)(Arow + k);
            v2f b = *(const v2f*)(Brow + k);
            acc = __builtin_amdgcn_wmma_f32_16x16x4_f32(
                false, a, false, b, (short)0, acc, false, false);
        }
    } else {
        const float* Bcol = B + (size_t)(col0 + mlocal);
#pragma unroll 4
        for (int k = 0; k < K; k += 4) {
            const int kb = k + 2 * khalf;
            v2f a = *(const v2f*)(Arow + k);
            v2f b;
            b.x = Bcol[(size_t)kb * N];
            b.y = Bcol[(size_t)(kb + 1) * N];
            acc = __builtin_amdgcn_wmma_f32_16x16x4_f32(
                false, a, false, b, (short)0, acc, false, false);
        }
    }

    const float bc = BIAS ? bias[col0 + mlocal] : 0.0f;
#pragma unroll
    for (int r = 0; r < 8; ++r) {
        const int row = row0 + r + 8 * khalf;
        C[(size_t)row * N + col0 + mlocal] = acc[r] + bc;
    }
}

// ---------------------------------------------------------------------------
// cvec[b] = sum_a bq[a] * dec_q[b,a]
// ---------------------------------------------------------------------------
__launch_bounds__(256)
__global__ void dot_bias_kernel(const float* __restrict__ dec_q,
                                const float* __restrict__ bq,
                                float* __restrict__ cvec, int A) {
    const int b   = blockIdx.x;
    const int tid = threadIdx.x;
    const float* dq = dec_q + (size_t)b * A;
    float s = 0.0f;
    for (int a = tid; a < A; a += 256) s += dq[a] * bq[a];
    __shared__ float red[256];
    red[tid] = s;
    __syncthreads();
    for (int st = 128; st > 0; st >>= 1) {
        if (tid < st) red[tid] += red[tid + st];
        __syncthreads();
    }
    if (tid == 0) cvec[b] = red[0];
}

// ---------------------------------------------------------------------------
// Fused per-batch attention core. One workgroup (16 waves) per batch b:
//  pass A: energy[l] = tanh(x[b,l,:] . u[b,:] + c[b])      (streams x_b: 1.6MB)
//  softmax over L in LDS (wave 0)
//  pass B: xbar[b,e] = sum_l w[l] * x[b,l,e]               (re-read hits L2)
// ---------------------------------------------------------------------------
#define ATT_THREADS 512
__launch_bounds__(ATT_THREADS)
__global__ void attn_fused_kernel(const float* __restrict__ x,   // [B,L,E]
                                  const float* __restrict__ u,   // [B,E]
                                  const float* __restrict__ cvec,// [B]
                                  float* __restrict__ xbar,      // [B,E]
                                  int L, int E) {
    const int b    = blockIdx.x;
    const int tid  = threadIdx.x;
    const int lane = tid & 31;
    const int wave = tid >> 5;
    const int nwav = ATT_THREADS >> 5;   // 16 waves

    __shared__ float sW[256];            // energies -> weights (L <= 256)

    const float* xb = x + (size_t)b * L * E;
    const float* ub = u + (size_t)b * E;
    const float  cb = cvec[b];

    // ---- pass A: energies (one row per wave at a time) ----
    for (int l = wave; l < L; l += nwav) {
        const float* row = xb + (size_t)l * E;
        float s = 0.0f;
        for (int e = lane * 4; e < E; e += 32 * 4) {
            float4 xv = *(const float4*)(row + e);
            float4 uv = *(const float4*)(ub + e);
            s += xv.x * uv.x + xv.y * uv.y + xv.z * uv.z + xv.w * uv.w;
        }
#pragma unroll
        for (int off = 16; off > 0; off >>= 1) s += __shfl_down(s, off, 32);
        if (lane == 0) sW[l] = tanhf(s + cb);
    }
    __syncthreads();

    // ---- softmax over L (wave 0 only; L=196 is tiny) ----
    if (wave == 0) {
        float m = -3.4e38f;
        for (int l = lane; l < L; l += 32) m = fmaxf(m, sW[l]);
#pragma unroll
        for (int off = 16; off > 0; off >>= 1) m = fmaxf(m, __shfl_xor(m, off, 32));
        float ssum = 0.0f;
        for (int l = lane; l < L; l += 32) {
            float ev = __expf(sW[l] - m);
            sW[l] = ev;
            ssum += ev;
        }
#pragma unroll
        for (int off = 16; off > 0; off >>= 1) ssum += __shfl_xor(ssum, off, 32);
        const float inv = 1.0f / ssum;
        for (int l = lane; l < L; l += 32) sW[l] *= inv;
    }
    __syncthreads();

    // ---- pass B: weighted sum over L; each thread owns one float4 of E ----
    const int e4 = tid;                  // E/4 == 512 == blockDim
    if (e4 * 4 < E) {
        float4 acc = {0.f, 0.f, 0.f, 0.f};
        const float* col = xb + (size_t)e4 * 4;
        for (int l = 0; l < L; ++l) {
            const float wl = sW[l];
            float4 xv = *(const float4*)(col + (size_t)l * E);
            acc.x += wl * xv.x;
            acc.y += wl * xv.y;
            acc.z += wl * xv.z;
            acc.w += wl * xv.w;
        }
        *(float4*)(xbar + (size_t)b * E + (size_t)e4 * 4) = acc;
    }
}

extern "C" void kernel_launch(void* const* d_in, const int* in_sizes, int n_in,
                              void* d_out, int out_size, void* d_ws, size_t ws_size,
                              hipStream_t stream) {
    const float* enc = (const float*)d_in[0]; // [B,L,E]
    const float* dh  = (const float*)d_in[1]; // [B,D]
    const float* Wq  = (const float*)d_in[2]; // [A,E]
    const float* bq  = (const float*)d_in[3]; // [A]
    const float* Wv  = (const float*)d_in[4]; // [A,E]
    const float* bv  = (const float*)d_in[5]; // [A]
    const float* Wd  = (const float*)d_in[6]; // [A,D]
    const float* bd  = (const float*)d_in[7]; // [A]
    float* out = (float*)d_out;               // [B,A]

    const int B = 128, L = 196, E = 2048, D = 1024, A = 1024;

    // workspace layout (floats); ~2.62 MB total
    float* ws    = (float*)d_ws;
    float* dec_q = ws;                    // B*A   = 131072
    float* u     = dec_q + (size_t)B * A; // B*E   = 262144
    float* cvec  = u + (size_t)B * E;     // B     = 128
    float* xbar  = cvec + B;              // B*E   = 262144

    // 1) dec_q = dh @ Wd^T + bd        (M=B, N=A, K=D), B transposed
    {
        dim3 grid(A / 16, B / 16);
        wmma_gemm_f32<true, true><<<grid, 32, 0, stream>>>(dh, Wd, bd, dec_q, B, A, D);
    }
    // 2) u = dec_q @ Wq                (M=B, N=E, K=A), B natural [K,N]
    {
        dim3 grid(E / 16, B / 16);
        wmma_gemm_f32<false, false><<<grid, 32, 0, stream>>>(dec_q, Wq, nullptr, u, B, E, A);
    }
    // 3) cvec[b] = bq . dec_q[b]
    dot_bias_kernel<<<B, 256, 0, stream>>>(dec_q, bq, cvec, A);

    // 4) fused energy/tanh/softmax/weighted-sum  (HBM-bound: one pass over enc)
    attn_fused_kernel<<<B, ATT_THREADS, 0, stream>>>(enc, u, cvec, xbar, L, E);

    // 5) out = xbar @ Wv^T + bv        (M=B, N=A, K=E), B transposed
    {
        dim3 grid(A / 16, B / 16);
        wmma_gemm_f32<true, true><<<grid, 32, 0, stream>>>(xbar, Wv, bv, out, B, A, E);
    }
}